// LGCN_Encoder_74646531604946
// MI455X (gfx1250) — compile-verified
//
#include <hip/hip_runtime.h>
#include <math.h>

#define EMB 64
#define GAMMA 0.5f
#define EPSV 1e-12f

// ---------------------------------------------------------------------------
// PROBE (first in file so the disasm snippet shows it): verifies that an
// agent-scope relaxed f32 fetch_add lowers to hardware global_atomic_add_f32
// (no-return form) rather than a CAS loop. Never launched.
// ---------------------------------------------------------------------------
__global__ void k_atom_probe(float* p, float v) {
  __hip_atomic_fetch_add(p, v, __ATOMIC_RELAXED, __HIP_MEMORY_SCOPE_AGENT);
}

// wave-uniform broadcast of lane j's value -> SGPR (no DS traffic)
__device__ __forceinline__ int bcast_i(int v, int j) {
  return __builtin_amdgcn_readlane(v, j);
}
__device__ __forceinline__ float bcast_f(float v, int j) {
  return __uint_as_float(__builtin_amdgcn_readlane(__float_as_uint(v), j));
}

// ---------------------------------------------------------------------------
// COO SpMM scatter: y[r,:] += v * xn[c,:] for each nonzero.
// Each wave grabs a chunk of 32 edges: lanes load metadata coalesced, then
// each edge is broadcast to SGPRs via v_readlane (loop index j is wave-
// uniform). All 32 lanes cooperate on its 64 floats: float2/lane b64 gather
// (saddr form, scalar base = xn + c*64) + 2x global_atomic_add_f32/lane.
// Both xn (38 MB) and y (38 MB) live in the 192 MB L2, so gathers hit L2 and
// atomics resolve at the L2 atomic units. Edge index math is 32-bit (nnz=8M); row offsets are size_t.
// ---------------------------------------------------------------------------
__global__ void k_scatter(const int* __restrict__ rows,
                          const int* __restrict__ cols,
                          const float* __restrict__ vals,
                          const float* __restrict__ xn,
                          float* __restrict__ y,
                          int nnz) {
  const int lane    = threadIdx.x & 31;
  const int gwave   = (blockIdx.x * blockDim.x + threadIdx.x) >> 5;
  const int nwaves  = (gridDim.x * blockDim.x) >> 5;
  const int stride  = nwaves * 32;
  const int laneoff = lane * 2;   // float2 element offset within a row

  for (int base = gwave * 32; base < nnz; base += stride) {
    // prefetch next metadata chunk (-> global_prefetch_b8)
    if (base + stride < nnz) {
      __builtin_prefetch(rows + base + stride + lane, 0, 0);
      __builtin_prefetch(cols + base + stride + lane, 0, 0);
      __builtin_prefetch(vals + base + stride + lane, 0, 0);
    }

    const int e = base + lane;
    int   r = 0, c = 0;
    float v = 0.0f;
    if (e < nnz) {
      r = rows[e];
      c = cols[e];
      v = vals[e];
    }

    if (base + 32 <= nnz) {
      // full chunk: branch-free, unrollable
#pragma unroll 4
      for (int j = 0; j < 32; ++j) {
        const int   rj = bcast_i(r, j);           // SGPR
        const int   cj = bcast_i(c, j);           // SGPR
        const float vj = bcast_f(v, j);           // SGPR

        const float2 xv = *(const float2*)(xn + (size_t)cj * EMB + laneoff);
        float* dst = y + (size_t)rj * EMB + laneoff;
        __hip_atomic_fetch_add(dst + 0, vj * xv.x, __ATOMIC_RELAXED,
                               __HIP_MEMORY_SCOPE_AGENT);
        __hip_atomic_fetch_add(dst + 1, vj * xv.y, __ATOMIC_RELAXED,
                               __HIP_MEMORY_SCOPE_AGENT);
      }
    } else {
      const int cnt = nnz - base;
      for (int j = 0; j < cnt; ++j) {
        const int   rj = bcast_i(r, j);
        const int   cj = bcast_i(c, j);
        const float vj = bcast_f(v, j);

        const float2 xv = *(const float2*)(xn + (size_t)cj * EMB + laneoff);
        float* dst = y + (size_t)rj * EMB + laneoff;
        __hip_atomic_fetch_add(dst + 0, vj * xv.x, __ATOMIC_RELAXED,
                               __HIP_MEMORY_SCOPE_AGENT);
        __hip_atomic_fetch_add(dst + 1, vj * xv.y, __ATOMIC_RELAXED,
                               __HIP_MEMORY_SCOPE_AGENT);
      }
    }
  }
}

// ---------------------------------------------------------------------------
// wave32 cross-lane sum (CDNA5 is wave32-only; never hardcode 64)
// ---------------------------------------------------------------------------
__device__ __forceinline__ float wave_reduce_sum(float v) {
#pragma unroll
  for (int off = 16; off > 0; off >>= 1) v += __shfl_xor(v, off, 32);
  return v;
}

// ---------------------------------------------------------------------------
// Layer 0: normalize rows of [user_emb; item_emb] into xn, zero y and prop.
// One wave per row, each lane owns a float2 (32 lanes * 2 = 64 elems).
// ---------------------------------------------------------------------------
__global__ void k_norm_first(const float* __restrict__ ue,
                             const float* __restrict__ ie,
                             float* __restrict__ xn,
                             float* __restrict__ y,
                             float* __restrict__ prop,
                             int user_num, int n) {
  const int gtid = blockIdx.x * blockDim.x + threadIdx.x;
  const int row  = gtid >> 5;
  const int lane = gtid & 31;
  if (row >= n) return;

  const float* src = (row < user_num) ? (ue + (size_t)row * EMB)
                                      : (ie + (size_t)(row - user_num) * EMB);
  float2 a = *(const float2*)(src + lane * 2);

  float ss = wave_reduce_sum(a.x * a.x + a.y * a.y);
  float s  = 1.0f / (sqrtf(ss) + EPSV);

  const size_t o = (size_t)row * EMB + (size_t)lane * 2;
  float2 z = make_float2(0.0f, 0.0f);
  *(float2*)(xn + o)   = make_float2(a.x * s, a.y * s);
  *(float2*)(y + o)    = z;
  *(float2*)(prop + o) = z;
}

// ---------------------------------------------------------------------------
// Layers 1..: prop += y; xn = y / ||y||; y = 0.
// Each wave owns one full row, so the read->zero of y has no cross-wave hazard.
// ---------------------------------------------------------------------------
__global__ void k_norm_accum(float* __restrict__ xn,
                             float* __restrict__ y,
                             float* __restrict__ prop,
                             int n) {
  const int gtid = blockIdx.x * blockDim.x + threadIdx.x;
  const int row  = gtid >> 5;
  const int lane = gtid & 31;
  if (row >= n) return;

  const size_t o = (size_t)row * EMB + (size_t)lane * 2;
  float2 a = *(const float2*)(y + o);
  float2 p = *(const float2*)(prop + o);
  p.x += a.x; p.y += a.y;
  *(float2*)(prop + o) = p;

  float ss = wave_reduce_sum(a.x * a.x + a.y * a.y);
  float s  = 1.0f / (sqrtf(ss) + EPSV);

  *(float2*)(xn + o) = make_float2(a.x * s, a.y * s);
  *(float2*)(y + o)  = make_float2(0.0f, 0.0f);
}

// ---------------------------------------------------------------------------
// out = GAMMA * ego0 + (1-GAMMA) * (prop + y) / N_LAYERS
// (after the 3rd scatter, prop holds ego1+ego2 and y holds ego3)
// ---------------------------------------------------------------------------
__global__ void k_final(const float* __restrict__ ue,
                        const float* __restrict__ ie,
                        const float* __restrict__ prop,
                        const float* __restrict__ y,
                        float* __restrict__ out,
                        int user_num, int n) {
  const int gtid = blockIdx.x * blockDim.x + threadIdx.x;
  const int row  = gtid >> 5;
  const int lane = gtid & 31;
  if (row >= n) return;

  const float* src = (row < user_num) ? (ue + (size_t)row * EMB)
                                      : (ie + (size_t)(row - user_num) * EMB);
  const size_t o = (size_t)row * EMB + (size_t)lane * 2;

  float2 e0 = *(const float2*)(src + lane * 2);
  float2 p  = *(const float2*)(prop + o);
  float2 yy = *(const float2*)(y + o);

  const float w = (1.0f - GAMMA) / 3.0f;  // N_LAYERS = 3
  float2 r;
  r.x = GAMMA * e0.x + w * (p.x + yy.x);
  r.y = GAMMA * e0.y + w * (p.y + yy.y);
  *(float2*)(out + o) = r;
}

// ---------------------------------------------------------------------------
// Host launcher
// ---------------------------------------------------------------------------
extern "C" void kernel_launch(void* const* d_in, const int* in_sizes, int n_in,
                              void* d_out, int out_size, void* d_ws, size_t ws_size,
                              hipStream_t stream) {
  const float* ue   = (const float*)d_in[0];
  const float* ie   = (const float*)d_in[1];
  const int*   rows = (const int*)d_in[2];
  const int*   cols = (const int*)d_in[3];
  const float* vals = (const float*)d_in[4];

  const int user_num = in_sizes[0] / EMB;
  const int item_num = in_sizes[1] / EMB;
  const int n        = user_num + item_num;
  const int nnz      = in_sizes[2];

  float* xn   = (float*)d_ws;                  // normalized source  [n*64]
  float* y    = xn + (size_t)n * EMB;          // scatter target     [n*64]
  float* prop = y + (size_t)n * EMB;           // running prop sum   [n*64]
  float* out  = (float*)d_out;

  const dim3 blk(256);                          // 8 waves / block (wave32)
  const int rowBlocks  = (n + 7) / 8;           // one wave per row
  const int chunks     = (nnz + 31) / 32;       // 32 edges per wave
  const int scatBlocks = (chunks + 7) / 8;

  // Layer 1
  k_norm_first<<<rowBlocks, blk, 0, stream>>>(ue, ie, xn, y, prop, user_num, n);
  k_scatter<<<scatBlocks, blk, 0, stream>>>(rows, cols, vals, xn, y, nnz);
  // Layer 2
  k_norm_accum<<<rowBlocks, blk, 0, stream>>>(xn, y, prop, n);
  k_scatter<<<scatBlocks, blk, 0, stream>>>(rows, cols, vals, xn, y, nnz);
  // Layer 3
  k_norm_accum<<<rowBlocks, blk, 0, stream>>>(xn, y, prop, n);
  k_scatter<<<scatBlocks, blk, 0, stream>>>(rows, cols, vals, xn, y, nnz);
  // Combine
  k_final<<<rowBlocks, blk, 0, stream>>>(ue, ie, prop, y, out, user_num, n);
}